// GIN_29403346109051
// MI455X (gfx1250) — compile-verified
//
#include <hip/hip_runtime.h>

// GIN (3 layers) for MI455X / gfx1250, fp32 end-to-end.
//  - edge scatter: wave-per-edge, float4 gather, hw f32 atomics
//  - GEMMs: V_WMMA_F32_16X16X4_F32; W pre-swizzled in global so LDS staging is
//    a contiguous copy, double-buffered via GLOBAL_LOAD_ASYNC_TO_LDS_B128
//    (ASYNCcnt) when the toolchain exposes it, float4 ds-store fallback else.
//  - BN: fused column-stat accumulation + tiny finalize kernel

typedef __attribute__((ext_vector_type(2))) float v2f;
typedef __attribute__((ext_vector_type(8))) float v8f;
typedef int v4i __attribute__((vector_size(16)));
typedef __attribute__((address_space(1))) v4i* gv4i_ptr;  // global v4i*
typedef __attribute__((address_space(3))) v4i* lv4i_ptr;  // LDS v4i*

#define N_NODES 40000
#define N_EDGES 640000
#define D 128
#define NLAYERS 3
#define BN_EPS 1e-5f

#define BM 64                 // rows per block in GEMM
#define XPAD 132              // padded LDS row stride -> conflict-free A-frag loads
#define KC 16                 // K-chunk staged in LDS per pass
#define NCHUNK (D / KC)       // 8
#define CHUNK_FLOATS (KC * D) // 2048 floats = 8KB per buffer

#if __has_builtin(__builtin_amdgcn_global_load_async_to_lds_b128)
#define HAVE_ASYNC_LDS 1
#else
#define HAVE_ASYNC_LDS 0
#endif

__device__ __forceinline__ void wait_async_lds() {
#if HAVE_ASYNC_LDS
#if __has_builtin(__builtin_amdgcn_s_wait_asynccnt)
    __builtin_amdgcn_s_wait_asynccnt(0);
#else
    asm volatile("s_wait_asynccnt 0" ::: "memory");
#endif
#endif
}

// issue (async) or perform (sync fallback) copy of one contiguous W chunk
__device__ __forceinline__ void stage_w_chunk(const float* __restrict__ src,
                                              float* __restrict__ dstLds,
                                              int tid) {
#if HAVE_ASYNC_LDS
    for (int i = tid; i < CHUNK_FLOATS / 4; i += 128) {
        __builtin_amdgcn_global_load_async_to_lds_b128(
            (gv4i_ptr)(src + i * 4),
            (lv4i_ptr)(dstLds + i * 4),
            0, 0);
    }
#else
    for (int i = tid; i < CHUNK_FLOATS / 4; i += 128) {
        *(float4*)(dstLds + i * 4) = *(const float4*)(src + i * 4);
    }
#endif
}

// ---------------------------------------------------------------- utilities
__global__ __launch_bounds__(256) void zero_kernel(float* __restrict__ p, int n) {
    int i = blockIdx.x * blockDim.x + threadIdx.x;
    if (i < n) p[i] = 0.0f;
}

// pair-swizzle W into global scratch so each lane's B-fragment
// {W[k][n], W[k+1][n]} is contiguous (one b64) and the LDS image is a
// contiguous copy of global memory (async-DMA friendly).
__global__ __launch_bounds__(256) void swizzle_W(const float* __restrict__ W,
                                                 float* __restrict__ Wsw) {
    int idx = blockIdx.x * blockDim.x + threadIdx.x;  // 0 .. D*D-1
    int k = idx >> 7;
    int n = idx & 127;
    Wsw[(k >> 1) * 256 + n * 2 + (k & 1)] = W[idx];
}

// ------------------------------------------------------------- edge scatter
// one wave (32 lanes) per edge; lane handles 4 consecutive features
__global__ __launch_bounds__(256) void edge_scatter(
    const float* __restrict__ h, const int* __restrict__ src,
    const int* __restrict__ dst, const float* __restrict__ ew,
    float* __restrict__ agg)
{
    int gid  = blockIdx.x * blockDim.x + threadIdx.x;
    int e    = gid >> 5;
    int lane = gid & 31;
    if (e >= N_EDGES) return;
    int   s = src[e];
    int   d = dst[e];
    float w = ew[e];
    float4 v = *(const float4*)(h + (size_t)s * D + lane * 4);
    float* ap = agg + (size_t)d * D + lane * 4;
    unsafeAtomicAdd(ap + 0, w * v.x);
    unsafeAtomicAdd(ap + 1, w * v.y);
    unsafeAtomicAdd(ap + 2, w * v.z);
    unsafeAtomicAdd(ap + 3, w * v.w);
}

// ---------------------------------------------------------------- WMMA GEMM
// mode 0:  X = (1+eps)*h + agg ; Y = X @ W + b             (stats of Y)
// mode 1:  X = relu(scale*y + shift) ; U = relu(X @ W + b) (stats of U)
// 128 threads = 4 waves; block computes 64 rows x 128 cols.
__global__ __launch_bounds__(128) void gin_gemm(
    const float* in,                       // h (mode0) or y (mode1)
    const float* __restrict__ agg,         // mode0 only
    const float* __restrict__ Wsw,         // pair-swizzled weights [D*D]
    const float* __restrict__ bias,        // [D]
    const float* __restrict__ epsp, int layer,
    const float* __restrict__ scale,       // mode1 only
    const float* __restrict__ shift,       // mode1 only
    float* out,                            // [N][D]
    float* __restrict__ colsum, float* __restrict__ colsumsq,
    int mode)
{
    __shared__ float sWc[2][CHUNK_FLOATS]; // double-buffered W chunks (2x8KB)
    __shared__ float sX[BM * XPAD];        // 64 * 132 floats (33KB)
    __shared__ float sBias[D];
    __shared__ float sScale[D];
    __shared__ float sShift[D];

    const int tid  = threadIdx.x;
    const int wave = tid >> 5;
    const int lane = tid & 31;
    const int half = lane >> 4;   // selects K-pair (A/B frags) and row-half (D)
    const int l16  = lane & 15;
    const int row0 = blockIdx.x * BM;

    // kick off chunk 0 of W while we stage X
    stage_w_chunk(Wsw, &sWc[0][0], tid);

    if (tid < D) {
        sBias[tid] = bias[tid];
        if (mode == 1) { sScale[tid] = scale[tid]; sShift[tid] = shift[tid]; }
    }
    float ce = 0.0f;
    if (mode == 0) ce = 1.0f + epsp[layer];
    __syncthreads();   // sScale/sShift ready for the X staging below

    // stage X tile (with fused pre-op) into LDS, float4 at a time
    for (int idx = tid; idx < BM * (D / 4); idx += blockDim.x) {
        int r  = idx >> 5;            // / (D/4)
        int c  = (idx & 31) * 4;
        size_t g = (size_t)(row0 + r) * D + c;
        float* px = &sX[r * XPAD + c];
        if (mode == 0) {
            float4 hv = *(const float4*)(in + g);
            float4 av = *(const float4*)(agg + g);
            px[0] = ce * hv.x + av.x;
            px[1] = ce * hv.y + av.y;
            px[2] = ce * hv.z + av.z;
            px[3] = ce * hv.w + av.w;
        } else {
            float4 yv = *(const float4*)(in + g);
            float t0 = sScale[c + 0] * yv.x + sShift[c + 0];
            float t1 = sScale[c + 1] * yv.y + sShift[c + 1];
            float t2 = sScale[c + 2] * yv.z + sShift[c + 2];
            float t3 = sScale[c + 3] * yv.w + sShift[c + 3];
            px[0] = t0 > 0.0f ? t0 : 0.0f;
            px[1] = t1 > 0.0f ? t1 : 0.0f;
            px[2] = t2 > 0.0f ? t2 : 0.0f;
            px[3] = t3 > 0.0f ? t3 : 0.0f;
        }
    }

    // accumulators: wave covers rows [wave*16, wave*16+16), 8 col-tiles of 16
    v8f acc[8];
#pragma unroll
    for (int t = 0; t < 8; ++t) {
        float bv = sBias[t * 16 + l16];
        v8f a;
#pragma unroll
        for (int i = 0; i < 8; ++i) a[i] = bv;
        acc[t] = a;
    }
    const int mloc = wave * 16 + l16;

#pragma unroll 2
    for (int kci = 0; kci < NCHUNK; ++kci) {
        const int cur = kci & 1;
        wait_async_lds();       // my async writes to sWc[cur] complete
        __syncthreads();        // everyone's writes done; prev reads of buf cur^1 done
        if (kci + 1 < NCHUNK)   // overlap next chunk's DMA with this chunk's WMMAs
            stage_w_chunk(Wsw + (size_t)(kci + 1) * CHUNK_FLOATS, &sWc[cur ^ 1][0], tid);

#pragma unroll
        for (int k0 = 0; k0 < KC; k0 += 4) {
            // A frag: {X[m][k], X[m][k+1]}, k = kci*KC + k0 + 2*half
            v2f afrag = *(const v2f*)&sX[mloc * XPAD + kci * KC + k0 + 2 * half];
            const float* wrow = &sWc[cur][((k0 >> 1) + half) * 256];
#pragma unroll
            for (int t = 0; t < 8; ++t) {
                // B frag: {W[k][n], W[k+1][n]}, n = t*16 + l16
                v2f bfrag = *(const v2f*)&wrow[(t * 16 + l16) * 2];
                acc[t] = __builtin_amdgcn_wmma_f32_16x16x4_f32(
                    false, afrag, false, bfrag, (short)0, acc[t], false, false);
            }
        }
    }

    // store D tiles + column statistics (sum / sumsq over rows)
#pragma unroll
    for (int t = 0; t < 8; ++t) {
        int col = t * 16 + l16;
        float s = 0.0f, sq = 0.0f;
#pragma unroll
        for (int r = 0; r < 8; ++r) {
            float v = acc[t][r];
            if (mode == 1) v = v > 0.0f ? v : 0.0f;
            int grow = row0 + wave * 16 + half * 8 + r;   // VGPR r -> row r + 8*half
            out[(size_t)grow * D + col] = v;
            s  += v;
            sq += v * v;
        }
        unsafeAtomicAdd(&colsum[col], s);
        unsafeAtomicAdd(&colsumsq[col], sq);
    }
}

// ------------------------------------------------------------ BN finalize
__global__ __launch_bounds__(128) void bn_finalize(
    const float* __restrict__ colsum, const float* __restrict__ colsumsq,
    const float* __restrict__ g, const float* __restrict__ bt,
    float* __restrict__ scale, float* __restrict__ shift)
{
    int c = threadIdx.x;
    float mean = colsum[c] * (1.0f / N_NODES);
    float var  = colsumsq[c] * (1.0f / N_NODES) - mean * mean;
    float rs   = rsqrtf(var + BN_EPS);
    float sc   = g[c] * rs;
    scale[c] = sc;
    shift[c] = bt[c] - mean * sc;
}

// --------------------------------------------------- outer BN + ReLU pass
__global__ __launch_bounds__(256) void bn_relu(
    const float* __restrict__ u, const float* __restrict__ scale,
    const float* __restrict__ shift, float* __restrict__ hout, int n)
{
    int i = blockIdx.x * blockDim.x + threadIdx.x;
    if (i < n) {
        int c = i & (D - 1);
        float v = scale[c] * u[i] + shift[c];
        hout[i] = v > 0.0f ? v : 0.0f;
    }
}

// ----------------------------------------------------------------- driver
extern "C" void kernel_launch(void* const* d_in, const int* in_sizes, int n_in,
                              void* d_out, int out_size, void* d_ws, size_t ws_size,
                              hipStream_t stream)
{
    const float* h0   = (const float*)d_in[0];
    const int*   src  = (const int*)d_in[1];
    const int*   dst  = (const int*)d_in[2];
    const float* ew   = (const float*)d_in[3];
    const float* epsp = (const float*)d_in[4];
    const float* W1   = (const float*)d_in[5];
    const float* b1   = (const float*)d_in[6];
    const float* g1   = (const float*)d_in[7];
    const float* bt1  = (const float*)d_in[8];
    const float* W2   = (const float*)d_in[9];
    const float* b2   = (const float*)d_in[10];
    const float* g2   = (const float*)d_in[11];
    const float* bt2  = (const float*)d_in[12];

    const size_t ND = (size_t)N_NODES * D;
    float* ws   = (float*)d_ws;
    float* agg  = ws;            // [N*D]
    float* ybuf = agg + ND;      // [N*D], reused in-place for GEMM2 output
    float* hA   = ybuf + ND;     // [N*D]
    float* hB   = hA + ND;       // [N*D]
    float* stats = hB + ND;      // 512 stat accumulators + 512 scale/shift
    float* colsum1   = stats;
    float* colsumsq1 = stats + 128;
    float* colsum2   = stats + 256;
    float* colsumsq2 = stats + 384;
    float* scale1 = stats + 512;
    float* shift1 = stats + 640;
    float* scale2 = stats + 768;
    float* shift2 = stats + 896;
    float* wsw1 = stats + 1024;         // [D*D] swizzled W1[l]
    float* wsw2 = wsw1 + (size_t)D * D; // [D*D] swizzled W2[l]

    const float* hin = h0;
    float* houts[NLAYERS] = { hA, hB, (float*)d_out };

    const int zeroBlocks = (int)((ND + 255) / 256);

    for (int l = 0; l < NLAYERS; ++l) {
        zero_kernel<<<zeroBlocks, 256, 0, stream>>>(agg, (int)ND);
        zero_kernel<<<2, 256, 0, stream>>>(stats, 512);
        swizzle_W<<<(D * D) / 256, 256, 0, stream>>>(W1 + (size_t)l * D * D, wsw1);
        swizzle_W<<<(D * D) / 256, 256, 0, stream>>>(W2 + (size_t)l * D * D, wsw2);

        edge_scatter<<<N_EDGES / 8, 256, 0, stream>>>(hin, src, dst, ew, agg);

        gin_gemm<<<N_NODES / BM, 128, 0, stream>>>(
            hin, agg, wsw1, b1 + (size_t)l * D,
            epsp, l, nullptr, nullptr,
            ybuf, colsum1, colsumsq1, 0);

        bn_finalize<<<1, 128, 0, stream>>>(colsum1, colsumsq1,
                                           g1 + (size_t)l * D, bt1 + (size_t)l * D,
                                           scale1, shift1);

        gin_gemm<<<N_NODES / BM, 128, 0, stream>>>(
            ybuf, nullptr, wsw2, b2 + (size_t)l * D,
            epsp, l, scale1, shift1,
            ybuf, colsum2, colsumsq2, 1);

        bn_finalize<<<1, 128, 0, stream>>>(colsum2, colsumsq2,
                                           g2 + (size_t)l * D, bt2 + (size_t)l * D,
                                           scale2, shift2);

        bn_relu<<<zeroBlocks, 256, 0, stream>>>(ybuf, scale2, shift2,
                                                houts[l], (int)ND);
        hin = houts[l];
    }
}